// GCN3_13065290515270
// MI455X (gfx1250) — compile-verified
//
#include <hip/hip_runtime.h>
#include <hip/hip_bf16.h>

typedef __attribute__((ext_vector_type(2))) float v2f;
typedef __attribute__((ext_vector_type(8))) float v8f;

#define HID 64
#define PD_EPS 1e-6f

// ---------------- degree / normalization ----------------
__global__ void k_deg_init(float* deg, int n) {
  int i = blockIdx.x * blockDim.x + threadIdx.x;
  if (i < n) deg[i] = 1.0f;  // self-loop contributes 1 to every node
}
__global__ void k_deg_count(const int* __restrict__ dst, float* deg, int e) {
  int i = blockIdx.x * blockDim.x + threadIdx.x;
  if (i < e) unsafeAtomicAdd(&deg[dst[i]], 1.0f);
}
__global__ void k_deg_fin(float* deg, int n) {
  int i = blockIdx.x * blockDim.x + threadIdx.x;
  if (i < n) deg[i] = rsqrtf(deg[i]);  // deg >= 1 always (self loops)
}

// ---------------- FP32 WMMA GEMM: OUT[N,64] = X[N,K] @ W[K,64] ----------------
// One block = 128 threads = 4 waves; block handles 16 rows, wave w handles
// output columns [16w, 16w+16). K stepped by 4 with V_WMMA_F32_16X16X4_F32.
// Requires nrows % 16 == 0 (100000 and 256 both satisfy this).
template <int K>
__global__ void k_gemm_n64_wmma(const float* __restrict__ X,
                                const float* __restrict__ W,
                                float* __restrict__ OUT) {
  const int lane  = threadIdx.x & 31;
  const int ntile = threadIdx.x >> 5;   // 0..3
  const int mtile = blockIdx.x;
  const int lo = lane & 15;
  const int hi = lane >> 4;             // 0 or 1

  // A 16x4 f32 layout: lanes 0-15 hold {K=0,K=1}, lanes 16-31 hold {K=2,K=3}
  const float* xrow = X + (size_t)(mtile * 16 + lo) * K + 2 * hi;
  // B 4x16 f32 layout: lanes 0-15 hold rows {0,1}, lanes 16-31 rows {2,3}, col = lo
  const float* wcol = W + (size_t)(2 * hi) * HID + (ntile * 16 + lo);

  v8f c = {};
#pragma unroll
  for (int k0 = 0; k0 < K; k0 += 4) {
    v2f a, b;
    a.x = xrow[k0];
    a.y = xrow[k0 + 1];
    b.x = wcol[(size_t)k0 * HID];
    b.y = wcol[(size_t)(k0 + 1) * HID];
    c = __builtin_amdgcn_wmma_f32_16x16x4_f32(
        /*neg_a=*/false, a, /*neg_b=*/false, b,
        /*c_mod=*/(short)0, c, /*reuse_a=*/false, /*reuse_b=*/false);
  }

  // C/D 16x16 f32: VGPR r -> M = r + 8*hi, N = lo
  float* o = OUT + (size_t)(mtile * 16 + 8 * hi) * HID + (ntile * 16 + lo);
#pragma unroll
  for (int r = 0; r < 8; ++r) o[(size_t)r * HID] = c[r];
}

// ---------------- aggregation ----------------
__global__ void k_selfloop(const float* __restrict__ dinv,
                           const float* __restrict__ h,
                           float* __restrict__ out, int n64) {
  int i = blockIdx.x * blockDim.x + threadIdx.x;
  if (i < n64) {
    float d = dinv[i >> 6];
    out[i] = d * d * h[i];
  }
}

// 256 threads = 4 edges/block, one feature per thread (coalesced 64-float rows)
__global__ void k_edge_scatter(const int* __restrict__ src,
                               const int* __restrict__ dst,
                               const float* __restrict__ dinv,
                               const float* __restrict__ h,
                               float* __restrict__ out, int nE) {
  int e = blockIdx.x * 4 + (threadIdx.x >> 6);
  int f = threadIdx.x & 63;
  if (e >= nE) return;
  int s = src[e], d = dst[e];
  float w = dinv[s] * dinv[d];
  unsafeAtomicAdd(&out[(size_t)d * HID + f], w * h[(size_t)s * HID + f]);
}

__global__ void k_bias_relu(float* x, const float* __restrict__ b, int n64,
                            int relu) {
  int i = blockIdx.x * blockDim.x + threadIdx.x;
  if (i < n64) {
    float v = x[i] + b[i & (HID - 1)];
    x[i] = relu ? fmaxf(v, 0.0f) : v;
  }
}

// ---------------- mean pool ----------------
__global__ void k_zero(float* p, int n) {
  int i = blockIdx.x * blockDim.x + threadIdx.x;
  if (i < n) p[i] = 0.0f;
}
__global__ void k_pool_add(const float* __restrict__ x,
                           const int* __restrict__ batch, float* pooled,
                           float* cnt, int n64) {
  int i = blockIdx.x * blockDim.x + threadIdx.x;
  if (i >= n64) return;
  int node = i >> 6, f = i & 63;
  int g = batch[node];
  unsafeAtomicAdd(&pooled[(size_t)g * HID + f], x[i]);
  if (f == 0) unsafeAtomicAdd(&cnt[g], 1.0f);
}
__global__ void k_pool_div(float* pooled, const float* __restrict__ cnt,
                           int g64) {
  int i = blockIdx.x * blockDim.x + threadIdx.x;
  if (i < g64) pooled[i] /= fmaxf(cnt[i >> 6], 1.0f);
}

// ---------------- pairwise distance ----------------
__global__ void k_pdist(const float* __restrict__ e1,
                        const float* __restrict__ e2, float* out) {
  int g = blockIdx.x, lane = threadIdx.x;
  float s = 0.0f;
  for (int f = lane; f < HID; f += 32) {
    float d = e1[(size_t)g * HID + f] - e2[(size_t)g * HID + f] + PD_EPS;
    s += d * d;
  }
  for (int off = 16; off > 0; off >>= 1) s += __shfl_down(s, off, 32);
  if (lane == 0) out[g] = sqrtf(s);
}

extern "C" void kernel_launch(void* const* d_in, const int* in_sizes, int n_in,
                              void* d_out, int out_size, void* d_ws,
                              size_t ws_size, hipStream_t stream) {
  const float* x1     = (const float*)d_in[0];
  const int*   ei1    = (const int*)d_in[1];
  const int*   batch1 = (const int*)d_in[2];
  const float* x2     = (const float*)d_in[3];
  const int*   ei2    = (const int*)d_in[4];
  const int*   batch2 = (const int*)d_in[5];
  const float* W1     = (const float*)d_in[6];
  const float* b1     = (const float*)d_in[7];
  const float* W2     = (const float*)d_in[8];
  const float* b2     = (const float*)d_in[9];
  const float* W3     = (const float*)d_in[10];
  const float* b3     = (const float*)d_in[11];
  const float* Wlin   = (const float*)d_in[12];
  const float* blin   = (const float*)d_in[13];

  const int N = in_sizes[2];      // 100000 nodes (batch length)
  const int E = in_sizes[1] / 2;  // 1.6M edges
  const int G = out_size;         // 256 graphs (output is [G])
  const int n64 = N * HID;
  const int g64 = G * HID;

  // workspace carve-out (floats): hbuf | obuf | dinv | pooled | cnt | emb1 | emb2
  float* hbuf   = (float*)d_ws;
  float* obuf   = hbuf + (size_t)n64;
  float* dinv   = obuf + (size_t)n64;
  float* pooled = dinv + N;
  float* cnt    = pooled + g64;     // contiguous after pooled
  float* emb1   = cnt + G;
  float* emb2   = emb1 + g64;

  const int TB = 256;
  auto tower = [&](const float* x, const int* ei, const int* batch,
                   float* emb) {
    const int* src = ei;
    const int* dst = ei + E;

    k_deg_init<<<(N + TB - 1) / TB, TB, 0, stream>>>(dinv, N);
    k_deg_count<<<(E + TB - 1) / TB, TB, 0, stream>>>(dst, dinv, E);
    k_deg_fin<<<(N + TB - 1) / TB, TB, 0, stream>>>(dinv, N);

    // layer 1: K = 128
    k_gemm_n64_wmma<128><<<N / 16, 128, 0, stream>>>(x, W1, hbuf);
    k_selfloop<<<(n64 + TB - 1) / TB, TB, 0, stream>>>(dinv, hbuf, obuf, n64);
    k_edge_scatter<<<(E + 3) / 4, TB, 0, stream>>>(src, dst, dinv, hbuf, obuf, E);
    k_bias_relu<<<(n64 + TB - 1) / TB, TB, 0, stream>>>(obuf, b1, n64, 1);

    // layer 2: K = 64
    k_gemm_n64_wmma<64><<<N / 16, 128, 0, stream>>>(obuf, W2, hbuf);
    k_selfloop<<<(n64 + TB - 1) / TB, TB, 0, stream>>>(dinv, hbuf, obuf, n64);
    k_edge_scatter<<<(E + 3) / 4, TB, 0, stream>>>(src, dst, dinv, hbuf, obuf, E);
    k_bias_relu<<<(n64 + TB - 1) / TB, TB, 0, stream>>>(obuf, b2, n64, 1);

    // layer 3: K = 64, no ReLU
    k_gemm_n64_wmma<64><<<N / 16, 128, 0, stream>>>(obuf, W3, hbuf);
    k_selfloop<<<(n64 + TB - 1) / TB, TB, 0, stream>>>(dinv, hbuf, obuf, n64);
    k_edge_scatter<<<(E + 3) / 4, TB, 0, stream>>>(src, dst, dinv, hbuf, obuf, E);
    k_bias_relu<<<(n64 + TB - 1) / TB, TB, 0, stream>>>(obuf, b3, n64, 0);

    // mean pool (pooled and cnt contiguous -> single zero pass)
    k_zero<<<(g64 + G + TB - 1) / TB, TB, 0, stream>>>(pooled, g64 + G);
    k_pool_add<<<(n64 + TB - 1) / TB, TB, 0, stream>>>(obuf, batch, pooled, cnt, n64);
    k_pool_div<<<(g64 + TB - 1) / TB, TB, 0, stream>>>(pooled, cnt, g64);

    // final linear: [G,64] @ [64,64]
    k_gemm_n64_wmma<64><<<G / 16, 128, 0, stream>>>(pooled, Wlin, emb);
    k_bias_relu<<<(g64 + TB - 1) / TB, TB, 0, stream>>>(emb, blin, g64, 0);
  };

  tower(x1, ei1, batch1, emb1);
  tower(x2, ei2, batch2, emb2);
  k_pdist<<<G, 32, 0, stream>>>(emb1, emb2, (float*)d_out);
}